// RoiProposal_88381837017327
// MI455X (gfx1250) — compile-verified
//
#include <hip/hip_runtime.h>
#include <cmath>
#include <stdint.h>

#ifndef __has_builtin
#define __has_builtin(x) 0
#endif

#define NANCH 9
#define PRE_NMS 6000
#define POST_NMS 300
#define SORT_N 8192            // next pow2 >= PRE_NMS for bitonic sort
#define NMS_TH 0.7f
#define MIN_SIZE 16.0f
#define IMG_W 3072.0f
#define IMG_H 2048.0f
#define XFORM_CLIP 4.135166556742356f   // log(1000/16)
#define CW 32                  // per-image counter words

// Classic py-faster-rcnn base anchors (base 16, ratios {0.5,1,2} x scales {8,16,32})
__constant__ float c_base[NANCH][4] = {
  { -84.f,  -40.f,  99.f,  55.f},
  {-176.f,  -88.f, 191.f, 103.f},
  {-360.f, -184.f, 375.f, 199.f},
  { -56.f,  -56.f,  71.f,  71.f},
  {-120.f, -120.f, 135.f, 135.f},
  {-248.f, -248.f, 263.f, 263.f},
  { -36.f,  -80.f,  51.f,  95.f},
  { -80.f, -168.f,  95.f, 183.f},
  {-168.f, -344.f, 183.f, 359.f}
};

// monotonic float -> uint key (larger key == larger float)
__device__ __forceinline__ unsigned ordkey(float f) {
  unsigned u = __float_as_uint(f);
  return (u & 0x80000000u) ? ~u : (u | 0x80000000u);
}

// ---------------------------------------------------------------------------
// Pass 1: fused softmax + bbox decode + clip + min-size filter -> boxes, keys
// ---------------------------------------------------------------------------
__global__ void decode_kernel(const float* __restrict__ cls,
                              const float* __restrict__ dlt,
                              float4* __restrict__ boxes,
                              unsigned* __restrict__ keys,
                              int N, int W) {
  int i = blockIdx.x * blockDim.x + threadIdx.x;
  int b = blockIdx.y;
  if (i >= N) return;
  int cell = i / NANCH, a = i - cell * NANCH;
  int yy = cell / W, xx = cell - yy * W;
  float sx = xx * 16.f, sy = yy * 16.f;

  float ax1 = c_base[a][0] + sx, ay1 = c_base[a][1] + sy;
  float ax2 = c_base[a][2] + sx, ay2 = c_base[a][3] + sy;
  float w  = ax2 - ax1 + 1.f,  h  = ay2 - ay1 + 1.f;
  float cx = ax1 + 0.5f * w,   cy = ay1 + 0.5f * h;

  size_t dbase = ((size_t)b * N + i) * 4;
  float dx = dlt[dbase + 0], dy = dlt[dbase + 1];
  float dw = fminf(dlt[dbase + 2], XFORM_CLIP);
  float dh = fminf(dlt[dbase + 3], XFORM_CLIP);

  float pcx = dx * w + cx, pcy = dy * h + cy;
  float pw  = __expf(dw) * w, ph = __expf(dh) * h;
  float x1 = fminf(fmaxf(pcx - 0.5f * pw, 0.f), IMG_W - 1.f);
  float y1 = fminf(fmaxf(pcy - 0.5f * ph, 0.f), IMG_H - 1.f);
  float x2 = fminf(fmaxf(pcx + 0.5f * pw, 0.f), IMG_W - 1.f);
  float y2 = fminf(fmaxf(pcy + 0.5f * ph, 0.f), IMG_H - 1.f);

  // 2-way softmax prob of class 1 == sigmoid(l1 - l0)
  size_t cbase = ((size_t)b * N + i) * 2;
  float l0 = cls[cbase + 0], l1 = cls[cbase + 1];
  float p = 1.f / (1.f + __expf(l0 - l1));

  bool ok = ((x2 - x1 + 1.f) >= MIN_SIZE) && ((y2 - y1 + 1.f) >= MIN_SIZE);
  float score = ok ? p : -1e30f;

  boxes[(size_t)b * N + i] = make_float4(x1, y1, x2, y2);
  keys [(size_t)b * N + i] = ordkey(score);
}

// ---------------------------------------------------------------------------
// Radix-select state init
// ---------------------------------------------------------------------------
__global__ void init_kernel(int* counters, int* hist) {
  int b = blockIdx.x, t = threadIdx.x;            // blockDim == 256
  if (t < CW) counters[b * CW + t] = (t == 1) ? PRE_NMS : 0;
  hist[b * 256 + t] = 0;
}

// Count current byte among prefix-matching keys
__global__ void hist_kernel(const unsigned* __restrict__ keys,
                            const int* __restrict__ counters,
                            int* __restrict__ hist, int N, int level) {
  __shared__ int lh[256];
  int t = threadIdx.x, b = blockIdx.y;
  lh[t] = 0;
  __syncthreads();
  unsigned prefix = (unsigned)counters[b * CW + 0];
  int i = blockIdx.x * blockDim.x + t;
  if (i < N) {
    unsigned key = keys[(size_t)b * N + i];
    bool part = (level == 0) || ((key >> (32 - 8 * level)) == prefix);
    if (part) atomicAdd(&lh[(key >> (24 - 8 * level)) & 0xFFu], 1);
  }
  __syncthreads();
  if (lh[t]) atomicAdd(&hist[b * 256 + t], lh[t]);
}

// Pick the bin containing the K-th largest; refine prefix; finalize at level 3
__global__ void scan_kernel(int* counters, int* hist, int level) {
  __shared__ int h[256];
  int b = blockIdx.x, t = threadIdx.x;            // blockDim == 256
  h[t] = hist[b * 256 + t];
  __syncthreads();
  if (t == 0) {
    int* c = &counters[b * CW];
    int remaining = c[1];
    int cum = 0, m = 0;
    for (int bin = 255; bin >= 0; --bin) {
      if (cum + h[bin] >= remaining) { m = bin; break; }
      cum += h[bin];
    }
    c[0] = (int)(((unsigned)c[0] << 8) | (unsigned)m);
    c[1] = remaining - cum;
    if (level == 3) {
      c[2] = c[0];   // exact threshold key
      c[3] = c[1];   // quota of threshold-equal keys to take
      c[4] = 0;      // compact slot counter
      c[5] = 0;      // equal-key counter
    }
  }
  __syncthreads();
  hist[b * 256 + t] = 0;
}

// ---------------------------------------------------------------------------
// Compact winners into (key<<32 | ~idx) composites (stable tie order)
// ---------------------------------------------------------------------------
__global__ void compact_kernel(const unsigned* __restrict__ keys,
                               int* counters,
                               unsigned long long* __restrict__ sel, int N) {
  int b = blockIdx.y;
  int i = blockIdx.x * blockDim.x + threadIdx.x;
  if (i >= N) return;
  int* c = &counters[b * CW];
  unsigned T = (unsigned)c[2];
  int quota = c[3];
  unsigned key = keys[(size_t)b * N + i];
  int slot = -1;
  if (key > T) {
    slot = atomicAdd(&c[4], 1);
  } else if (key == T) {
    int e = atomicAdd(&c[5], 1);
    if (e < quota) slot = atomicAdd(&c[4], 1);
  }
  if (slot >= 0 && slot < PRE_NMS)
    sel[(size_t)b * SORT_N + slot] =
        ((unsigned long long)key << 32) | (unsigned)(0xFFFFFFFFu - (unsigned)i);
}

// ---------------------------------------------------------------------------
// One workgroup/image: bitonic sort 8192 x u64 fully in LDS (64KB), then
// gather the 6000 winning boxes into contiguous score-order.
// ---------------------------------------------------------------------------
__global__ void __launch_bounds__(1024)
sort_kernel(const unsigned long long* __restrict__ sel,
            const int* __restrict__ counters,
            const float4* __restrict__ boxes,
            float4* __restrict__ sboxes,
            unsigned* __restrict__ skeys, int N) {
  __shared__ unsigned long long s[SORT_N];       // 64 KB LDS
  int b = blockIdx.x, t = threadIdx.x, bd = blockDim.x;
  int cnt = counters[b * CW + 4];
  if (cnt > PRE_NMS) cnt = PRE_NMS;
  for (int r = t; r < SORT_N; r += bd)
    s[r] = (r < cnt) ? sel[(size_t)b * SORT_N + r] : 0ULL;
  __syncthreads();
  for (int k = 2; k <= SORT_N; k <<= 1) {
    for (int j = k >> 1; j > 0; j >>= 1) {
      for (int p = t; p < SORT_N / 2; p += bd) {
        int i = 2 * p - (p & (j - 1));
        int ixj = i + j;
        unsigned long long a = s[i], c2 = s[ixj];
        bool desc = ((i & k) == 0);
        if (desc ? (a < c2) : (a > c2)) { s[i] = c2; s[ixj] = a; }
      }
      __syncthreads();
    }
  }
  for (int r = t; r < PRE_NMS; r += bd) {
    unsigned long long v = s[r];
    skeys[(size_t)b * PRE_NMS + r] = (unsigned)(v >> 32);
    float4 bx = make_float4(0.f, 0.f, 0.f, 0.f);
    if (v != 0ULL) {
      int idx = (int)(0xFFFFFFFFu - (unsigned)v);
      if (idx >= 0 && idx < N) bx = boxes[(size_t)b * N + idx];
    }
    sboxes[(size_t)b * PRE_NMS + r] = bx;
  }
}

// ---------------------------------------------------------------------------
// Greedy NMS per image. Boxes staged via CDNA5 async global->LDS (ASYNCcnt),
// keep-bits in LDS, uniform early-exit at 300 survivors. Emits output rows.
// ---------------------------------------------------------------------------
#if __has_builtin(__builtin_amdgcn_global_load_async_to_lds_b128)
#define HAVE_ASYNC_LDS 1
typedef int v4i_t __attribute__((ext_vector_type(4)));
typedef __attribute__((address_space(1))) v4i_t* gv4i_p;
typedef __attribute__((address_space(3))) v4i_t* lv4i_p;
#else
#define HAVE_ASYNC_LDS 0
#endif

__global__ void __launch_bounds__(1024)
nms_kernel(const float4* __restrict__ sboxes,
           const unsigned* __restrict__ skeys,
           float* __restrict__ out) {
  extern __shared__ char smem[];                 // PRE_NMS * 16 bytes of boxes
  float4* lb = (float4*)smem;
  __shared__ unsigned lkey[PRE_NMS];
  __shared__ unsigned keepw[(PRE_NMS + 31) / 32];
  int b = blockIdx.x, t = threadIdx.x, bd = blockDim.x;
  const float4* gsb = sboxes + (size_t)b * PRE_NMS;

  for (int r = t; r < PRE_NMS; r += bd) {
#if HAVE_ASYNC_LDS
    __builtin_amdgcn_global_load_async_to_lds_b128(
        (gv4i_p)(void*)(gsb + r),
        (lv4i_p)(void*)(lb + r), 0, 0);
#else
    lb[r] = gsb[r];
#endif
    lkey[r] = skeys[(size_t)b * PRE_NMS + r];
  }
  for (int r = t; r < (PRE_NMS + 31) / 32; r += bd) keepw[r] = 0xFFFFFFFFu;
#if HAVE_ASYNC_LDS
#if __has_builtin(__builtin_amdgcn_s_wait_asynccnt)
  __builtin_amdgcn_s_wait_asynccnt(0);
#else
  asm volatile("s_wait_asynccnt 0" ::: "memory");
#endif
#endif
  __syncthreads();

  const unsigned VALID = ordkey(-1e20f);
  int kept = 0;
  for (int i = 0; i < PRE_NMS; ++i) {
    bool keep_i = (keepw[i >> 5] >> (i & 31)) & 1u;   // uniform across block
    if (keep_i) {
      if (lkey[i] > VALID) {
        ++kept;
        if (t == 0) {
          float4 bi0 = lb[i];
          float* row = out + ((size_t)b * POST_NMS + (kept - 1)) * 5;
          row[0] = (float)b;
          row[1] = bi0.x; row[2] = bi0.y; row[3] = bi0.z; row[4] = bi0.w;
        }
        if (kept >= POST_NMS) break;                  // uniform exit
      }
      float4 bi = lb[i];
      float area_i = (bi.z - bi.x + 1.f) * (bi.w - bi.y + 1.f);
      for (int j = i + 1 + t; j < PRE_NMS; j += bd) {
        if ((keepw[j >> 5] >> (j & 31)) & 1u) {
          float4 bj = lb[j];
          float xx1 = fmaxf(bi.x, bj.x), yy1 = fmaxf(bi.y, bj.y);
          float xx2 = fminf(bi.z, bj.z), yy2 = fminf(bi.w, bj.w);
          float iw = fmaxf(xx2 - xx1 + 1.f, 0.f);
          float ih = fmaxf(yy2 - yy1 + 1.f, 0.f);
          float inter = iw * ih;
          float aj = (bj.z - bj.x + 1.f) * (bj.w - bj.y + 1.f);
          float iou = inter / (area_i + aj - inter);
          if (iou > NMS_TH) atomicAnd(&keepw[j >> 5], ~(1u << (j & 31)));
        }
      }
    }
    __syncthreads();
  }
  __syncthreads();
  // zero-fill unused output rows for this image
  int rem = (POST_NMS - kept) * 5;
  float* basep = out + ((size_t)b * POST_NMS + kept) * 5;
  for (int r = t; r < rem; r += bd) basep[r] = 0.f;
}

// ---------------------------------------------------------------------------
extern "C" void kernel_launch(void* const* d_in, const int* in_sizes, int n_in,
                              void* d_out, int out_size, void* d_ws, size_t ws_size,
                              hipStream_t stream) {
  const float* cls = (const float*)d_in[0];   // (B,H,W,18) fp32
  const float* dlt = (const float*)d_in[1];   // (B,H,W,36) fp32
  float* out = (float*)d_out;                 // (B*300, 5) fp32

  int B = out_size / (POST_NMS * 5);
  if (B < 1) B = 1;
  long long hw18 = (long long)in_sizes[0] / B;
  int HW = (int)(hw18 / 18);
  int N = HW * NANCH;
  // infer W from the 3:2 image aspect (2048x3072 @ stride 16 -> 128x192)
  int W = (int)(std::sqrt((double)HW * 1.5) + 0.5);
  if (W < 1) W = 1;
  while (W > 1 && (HW % W) != 0) --W;

  // carve workspace
  char* p = (char*)d_ws;
  auto carve = [&](size_t bytes) { char* r = p; p += (bytes + 255) & ~(size_t)255; return r; };
  float4*             boxes    = (float4*)            carve((size_t)B * N * sizeof(float4));
  unsigned*           keys     = (unsigned*)          carve((size_t)B * N * sizeof(unsigned));
  unsigned long long* sel      = (unsigned long long*)carve((size_t)B * SORT_N * 8);
  float4*             sboxes   = (float4*)            carve((size_t)B * PRE_NMS * sizeof(float4));
  unsigned*           skeys    = (unsigned*)          carve((size_t)B * PRE_NMS * sizeof(unsigned));
  int*                counters = (int*)               carve((size_t)B * CW * sizeof(int));
  int*                hist     = (int*)               carve((size_t)B * 256 * sizeof(int));

  dim3 gN((unsigned)((N + 255) / 256), (unsigned)B);

  decode_kernel<<<gN, 256, 0, stream>>>(cls, dlt, boxes, keys, N, W);
  init_kernel<<<B, 256, 0, stream>>>(counters, hist);
  for (int level = 0; level < 4; ++level) {
    hist_kernel<<<gN, 256, 0, stream>>>(keys, counters, hist, N, level);
    scan_kernel<<<B, 256, 0, stream>>>(counters, hist, level);
  }
  compact_kernel<<<gN, 256, 0, stream>>>(keys, counters, sel, N);
  sort_kernel<<<B, 1024, 0, stream>>>(sel, counters, boxes, sboxes, skeys, N);
  nms_kernel<<<B, 1024, PRE_NMS * sizeof(float4), stream>>>(sboxes, skeys, out);
}